// Decade_Weighted_Loss_60421599920187
// MI455X (gfx1250) — compile-verified
//
#include <hip/hip_runtime.h>
#include <hip/hip_bf16.h>

// Decade-weighted L2 loss, single pass.
//   S[b,bin] = sum of (p-t)^2 for samples of row b landing in bin
//   C[b,bin] = count of samples of row b in bin
//   loss = sqrt( (Σ S/C) / (#nonempty bins) )
// Memory-bound: 256 MiB streamed once ≈ 11 us at 23.3 TB/s HBM.
// Inputs are touched exactly once (>L2 capacity) -> non-temporal loads.

#define NBINS 64
#define NROWS 64
#define BLOCK 256
#define WAVES (BLOCK / 32)
#define CHUNK 8192   // elements per block (row-chunk)

typedef __attribute__((ext_vector_type(2))) float v2f;
typedef __attribute__((ext_vector_type(4))) float v4f;   // native vector for NT loads
typedef __attribute__((ext_vector_type(8))) float v8f;

__global__ void dwl_init(float* __restrict__ gS, unsigned* __restrict__ gC) {
  int i = blockIdx.x * blockDim.x + threadIdx.x;
  if (i < NROWS * NBINS) {
    gS[i] = 0.0f;
    gC[i] = 0u;
  }
}

__device__ __forceinline__ void dwl_acc(float pv, float tv,
                                        float* sS, unsigned* sC) {
  float d = pv - tv;
  int b = (int)fabsf(tv);           // floor of non-negative; eps irrelevant (<1 -> bin 0)
  b = b > (NBINS - 1) ? (NBINS - 1) : b;
  atomicAdd(&sS[b], d * d);         // ds_add_f32
  atomicAdd(&sC[b], 1u);            // ds_add_u32
}

__global__ __launch_bounds__(BLOCK) void dwl_hist(
    const float* __restrict__ yp, const float* __restrict__ yt,
    float* __restrict__ gS, unsigned* __restrict__ gC, int T) {
  __shared__ float    sS[WAVES][NBINS];
  __shared__ unsigned sC[WAVES][NBINS];

  const int wave = threadIdx.x >> 5;      // wave32
  const int lane = threadIdx.x & 31;

  for (int b = lane; b < NBINS; b += 32) {
    sS[wave][b] = 0.0f;
    sC[wave][b] = 0u;
  }
  __syncthreads();

  const int row     = blockIdx.y;
  const int rowBase = row * T;
  const int start   = rowBase + blockIdx.x * CHUNK;
  int stop = start + CHUNK;
  const int rowEnd = rowBase + T;
  if (stop > rowEnd) stop = rowEnd;

  // Vectorized main body: 128-bit non-temporal -> global_load_b128 th:NT.
  const int nvec = (stop - start) >> 2;
  const v4f* __restrict__ p4 = (const v4f*)(yp + start);
  const v4f* __restrict__ t4 = (const v4f*)(yt + start);
  for (int i = threadIdx.x; i < nvec; i += BLOCK) {
    v4f t = __builtin_nontemporal_load(&t4[i]);
    v4f p = __builtin_nontemporal_load(&p4[i]);
    dwl_acc(p.x, t.x, sS[wave], sC[wave]);
    dwl_acc(p.y, t.y, sS[wave], sC[wave]);
    dwl_acc(p.z, t.z, sS[wave], sC[wave]);
    dwl_acc(p.w, t.w, sS[wave], sC[wave]);
  }
  // Scalar tail (none for T = 2^19, kept for generality).
  for (int i = start + (nvec << 2) + threadIdx.x; i < stop; i += BLOCK) {
    float tv = __builtin_nontemporal_load(&yt[i]);
    float pv = __builtin_nontemporal_load(&yp[i]);
    dwl_acc(pv, tv, sS[wave], sC[wave]);
  }

  __syncthreads();

  // Combine wave-private histograms; one global atomic pair per bin per block.
  for (int b = threadIdx.x; b < NBINS; b += BLOCK) {
    float    s = 0.0f;
    unsigned c = 0u;
    #pragma unroll
    for (int w = 0; w < WAVES; ++w) {
      s += sS[w][b];
      c += sC[w][b];
    }
    if (c) {
      atomicAdd(&gS[row * NBINS + b], s);
      atomicAdd(&gC[row * NBINS + b], c);
    }
  }
}

// One wave (32 lanes, EXEC all ones). Per-lane partials over the 64x64 table,
// then cross-lane reduction on the matrix pipe:
//   A = (v, v) per lane, B = ones  =>  D[m,n] = 2*(v(m) + v(m+16))
// Per-lane sum of the 8 accumulator VGPRs + shfl_xor(16) gives 2*Σv; the
// factor 2 cancels in the ratio (ΣS/C) / (Σw).
__global__ __launch_bounds__(32) void dwl_final(
    const float* __restrict__ gS, const unsigned* __restrict__ gC,
    float* __restrict__ out) {
  const int lane = threadIdx.x;

  float r = 0.0f;  // Σ S/C over this lane's slice
  float w = 0.0f;  // # nonempty bins over this lane's slice
  for (int i = lane; i < NROWS * NBINS; i += 32) {
    unsigned c = gC[i];
    if (c) {
      r += gS[i] / (float)c;
      w += 1.0f;
    }
  }

  v2f ar;   ar[0] = r;    ar[1] = r;
  v2f aw;   aw[0] = w;    aw[1] = w;
  v2f ones; ones[0] = 1.0f; ones[1] = 1.0f;
  v8f cr = {};
  v8f cw = {};
  cr = __builtin_amdgcn_wmma_f32_16x16x4_f32(
      /*neg_a=*/false, ar, /*neg_b=*/false, ones,
      /*c_mod=*/(short)0, cr, /*reuse_a=*/false, /*reuse_b=*/false);
  cw = __builtin_amdgcn_wmma_f32_16x16x4_f32(
      /*neg_a=*/false, aw, /*neg_b=*/false, ones,
      /*c_mod=*/(short)0, cw, /*reuse_a=*/false, /*reuse_b=*/false);

  float pr = cr[0] + cr[1] + cr[2] + cr[3] + cr[4] + cr[5] + cr[6] + cr[7];
  float pw = cw[0] + cw[1] + cw[2] + cw[3] + cw[4] + cw[5] + cw[6] + cw[7];
  pr += __shfl_xor(pr, 16, 32);   // add the other half of rows (ds_bpermute)
  pw += __shfl_xor(pw, 16, 32);

  if (lane == 0) {
    out[0] = sqrtf(pr / pw);      // factor 2 from the WMMA trick cancels
  }
}

extern "C" void kernel_launch(void* const* d_in, const int* in_sizes, int n_in,
                              void* d_out, int out_size, void* d_ws, size_t ws_size,
                              hipStream_t stream) {
  const float* yp = (const float*)d_in[0];   // y_pred
  const float* yt = (const float*)d_in[1];   // y_true
  const int n = in_sizes[0];                 // B*T*1 = 33,554,432
  const int T = n / NROWS;

  float*    gS = (float*)d_ws;
  unsigned* gC = (unsigned*)((char*)d_ws + (size_t)NROWS * NBINS * sizeof(float));

  dwl_init<<<(NROWS * NBINS + 255) / 256, 256, 0, stream>>>(gS, gC);

  const int chunks = (T + CHUNK - 1) / CHUNK;   // 64 for T = 2^19
  dim3 grid(chunks, NROWS);
  dwl_hist<<<grid, BLOCK, 0, stream>>>(yp, yt, gS, gC, T);

  dwl_final<<<1, 32, 0, stream>>>(gS, gC, (float*)d_out);
}